// FederatedInvariantAlignment_54391465837022
// MI455X (gfx1250) — compile-verified
//
#include <hip/hip_runtime.h>
#include <stdint.h>

#define N_FULL 2048
#define K_CLI  8
#define NK_CLI 1024
#define TAU_C  0.5f
#define TAU_S  0.5f
#define GAMMA  1.0f
#define BLK    256

#define AS1 __attribute__((address_space(1)))
#define AS3 __attribute__((address_space(3)))

typedef int v4i __attribute__((__vector_size__(4 * sizeof(int))));

#if __has_builtin(__builtin_amdgcn_global_load_async_to_lds_b128)
#define HAVE_ASYNC_LDS 1
#endif

// ---------------------------------------------------------------------------
// Kernel 2 (defined first so its asm appears at the top of the disassembly):
// one output element per thread.
//   n, sum, cnt accumulated in k-order; values kept in registers so that
//   V = sum_k (v_k - mean)^2 / n matches the reference formula exactly.
// Column positions for this thread (32 contiguous bytes in the transposed
// pos table) are staged into LDS with the gfx1250 async-to-LDS path
// (global_load_async_to_lds_b128 + s_wait_asynccnt).
// ---------------------------------------------------------------------------
__global__ __launch_bounds__(BLK)
void fed_align_kernel(const float* __restrict__ adj,
                      const int*   __restrict__ pos,
                      float*       __restrict__ out) {
    __shared__ __align__(16) int posCol[BLK * K_CLI];

    const int tid = threadIdx.x;
    const int j   = blockIdx.x * BLK + tid;   // output column
    const int i   = blockIdx.y;               // output row

    // ---- stage pos[j*8 .. j*8+7] (32 bytes) into LDS -----------------------
#ifdef HAVE_ASYNC_LDS
    {
        const void* gsrc = (const void*)(pos + (size_t)j * K_CLI);
        void*       ldst = (void*)(&posCol[tid * K_CLI]);
        AS1 v4i* g = (AS1 v4i*)(uintptr_t)gsrc;
        AS3 v4i* l = (AS3 v4i*)(uint32_t)(uintptr_t)ldst;  // low 32 bits = LDS offset
        __builtin_amdgcn_global_load_async_to_lds_b128(g, l, 0,  0);
        __builtin_amdgcn_global_load_async_to_lds_b128(g, l, 16, 0);
#if __has_builtin(__builtin_amdgcn_s_wait_asynccnt)
        __builtin_amdgcn_s_wait_asynccnt(0);
#else
        asm volatile("s_wait_asynccnt 0x0" ::: "memory");
#endif
        asm volatile("" ::: "memory");  // fence LDS reads after the wait
    }
#else
    #pragma unroll
    for (int k = 0; k < K_CLI; ++k)
        posCol[tid * K_CLI + k] = pos[(size_t)j * K_CLI + k];
#endif
    // Each thread reads back only the LDS entries its own lane wrote, so no
    // cross-wave barrier is required.

    const int* prow = pos + (size_t)i * K_CLI;  // wave-uniform -> scalar loads

    float v[K_CLI];
    unsigned pm = 0;     // presence bitmask
    float sum = 0.0f;
    int   n   = 0;
    int   cnt = 0;

    #pragma unroll
    for (int k = 0; k < K_CLI; ++k) {
        const int a = prow[k];                    // row slot in client k (uniform)
        const int b = posCol[tid * K_CLI + k];    // col slot in client k
        float val = 0.0f;
        if ((a >= 0) && (b >= 0)) {
            const int lin = (((k << 10) + a) << 10) + b;  // (k*1024 + a)*1024 + b
            val = adj[lin];
            pm |= (1u << k);
            n  += 1;
            sum += val;
            cnt += (val > TAU_C) ? 1 : 0;
        }
        v[k] = val;
    }

    float res = 0.0f;
    if (n > 0) {
        const float fn   = (float)n;              // num_obs = max(n, 1e-5) == n
        const float mean = sum / fn;
        float V = 0.0f;
        #pragma unroll
        for (int k = 0; k < K_CLI; ++k) {
            if (pm & (1u << k)) {
                const float d = v[k] - mean;
                V += d * d;
            }
        }
        V /= fn;
        const float C = (float)cnt / fn;
        const float S = C * expf(-GAMMA * V);
        if (S > TAU_S) res = mean;                // * (num_obs > 1e-5) == 1 here
    }
    out[(size_t)i * N_FULL + j] = res;
}

// ---------------------------------------------------------------------------
// Kernel 1: build inverse index table, transposed layout pos[i*8 + k].
// pos[i*8+k] = a  if client_idx[k][a] == i, else -1.
// One block per client k; scatter through LDS (indices are unique per client).
// All trip counts are exact multiples of BLK -> fully unrolled, no predication.
// ---------------------------------------------------------------------------
__global__ __launch_bounds__(BLK)
void fed_build_pos_kernel(const int* __restrict__ idx, int* __restrict__ pos) {
    __shared__ int sp[N_FULL];
    const int k   = blockIdx.x;
    const int tid = threadIdx.x;

    #pragma unroll
    for (int u = 0; u < N_FULL / BLK; ++u)
        sp[tid + u * BLK] = -1;
    __syncthreads();

    #pragma unroll
    for (int u = 0; u < NK_CLI / BLK; ++u) {
        const int a = tid + u * BLK;
        sp[idx[k * NK_CLI + a]] = a;
    }
    __syncthreads();

    #pragma unroll
    for (int u = 0; u < N_FULL / BLK; ++u) {
        const int t = tid + u * BLK;
        pos[t * K_CLI + k] = sp[t];
    }
}

// ---------------------------------------------------------------------------
extern "C" void kernel_launch(void* const* d_in, const int* in_sizes, int n_in,
                              void* d_out, int out_size, void* d_ws, size_t ws_size,
                              hipStream_t stream) {
    const float* adj = (const float*)d_in[0];  // (8, 1024, 1024) f32
    const int*   idx = (const int*)  d_in[1];  // (8, 1024) i32
    float*       out = (float*)d_out;          // (2048, 2048) f32
    int*         pos = (int*)d_ws;             // (2048, 8) i32 inverse table (64 KB)

    fed_build_pos_kernel<<<K_CLI, BLK, 0, stream>>>(idx, pos);

    dim3 grid(N_FULL / BLK, N_FULL, 1);
    fed_align_kernel<<<grid, BLK, 0, stream>>>(adj, pos, out);
}